// HybridGraphModel_28913719837317
// MI455X (gfx1250) — compile-verified
//
#include <hip/hip_runtime.h>
#include <hip/hip_bf16.h>

typedef __attribute__((ext_vector_type(2))) float v2f;
typedef __attribute__((ext_vector_type(8))) float v8f;

// ---------------------------------------------------------------------------
// Utility: fill a float buffer with a constant (workspace init must be done
// by us every call; harness does not re-zero between replays).
// ---------------------------------------------------------------------------
__global__ void fill_kernel(float* __restrict__ p, float v, long long n) {
    long long gid = (long long)blockIdx.x * blockDim.x + threadIdx.x;
    if (gid < n) p[gid] = v;
}

// ---------------------------------------------------------------------------
// Degree: deg[dst] += 1 per edge (deg pre-filled with 1.0 for self-loops).
// ---------------------------------------------------------------------------
__global__ void deg_kernel(const int* __restrict__ dst, float* __restrict__ deg, int E) {
    int e = blockIdx.x * blockDim.x + threadIdx.x;
    if (e < E) atomicAdd(&deg[dst[e]], 1.0f);
}

__global__ void rsqrt_kernel(float* __restrict__ d, int N) {
    int i = blockIdx.x * blockDim.x + threadIdx.x;
    if (i < N) d[i] = rsqrtf(d[i]);
}

// ---------------------------------------------------------------------------
// WMMA GEMM: H[N,64] = act(X)[N,K] @ W[K,64], fp32 via V_WMMA_F32_16X16X4_F32.
// RELU_IN applies ReLU to X while staging into LDS (fuses previous layer's
// activation into this kernel's load phase).
// Block = 256 threads (8 wave32), covers 32 rows x 64 cols.
// Wave w: rowTile = w>>2 (16 rows), colTile = w&3 (16 cols).
// A frag (16x4 f32, 2 VGPR): lanes 0-15 -> K=k..k+1, lanes 16-31 -> K=k+2..k+3.
// B frag (4x16 f32, 2 VGPR): mirrored K split, N striped across lanes.
// C/D (16x16 f32, 8 VGPR): VGPR r -> M=r (lanes 0-15) / M=r+8 (lanes 16-31).
// ---------------------------------------------------------------------------
template <int K, bool RELU_IN>
__global__ __launch_bounds__(256) void gemm_wmma_kernel(
    const float* __restrict__ X, const float* __restrict__ W,
    float* __restrict__ H, int N) {
    // Row strides keep every row 16B-aligned (K+4 and 80 are multiples of 4)
    // and make the two half-waves of a wmma fragment read disjoint banks.
    __shared__ __align__(16) float Xs[32][K + 4];
    __shared__ __align__(16) float Ws[K][80];

    const int tid = threadIdx.x;
    const int rowBlock = blockIdx.x * 32;

    // Stage X tile (32 x K) into LDS, float4 at a time.
    for (int i = tid; i < 32 * (K / 4); i += 256) {
        int r = i / (K / 4), c4 = (i % (K / 4)) * 4;
        int grow = rowBlock + r;
        float4 v = make_float4(0.f, 0.f, 0.f, 0.f);
        if (grow < N) v = *(const float4*)(X + (size_t)grow * K + c4);
        if (RELU_IN) {
            v.x = fmaxf(v.x, 0.f); v.y = fmaxf(v.y, 0.f);
            v.z = fmaxf(v.z, 0.f); v.w = fmaxf(v.w, 0.f);
        }
        *(float4*)&Xs[r][c4] = v;
    }
    // Stage full W (K x 64) into LDS, float4 at a time.
    for (int i = tid; i < K * 16; i += 256) {
        int r = i >> 4, c4 = (i & 15) << 2;
        *(float4*)&Ws[r][c4] = *(const float4*)(W + (size_t)r * 64 + c4);
    }
    __syncthreads();

    const int lane = tid & 31;
    const int wave = tid >> 5;
    const int rowTile = wave >> 2;   // 0..1
    const int colTile = wave & 3;    // 0..3
    const int m  = lane & 15;
    const int hi = lane >> 4;        // 0: K pair {0,1}, 1: K pair {2,3}

    v8f acc = {};
#pragma unroll 8
    for (int k = 0; k < K; k += 4) {
        v2f a, b;
        a[0] = Xs[rowTile * 16 + m][k + 2 * hi + 0];
        a[1] = Xs[rowTile * 16 + m][k + 2 * hi + 1];
        b[0] = Ws[k + 2 * hi + 0][colTile * 16 + m];
        b[1] = Ws[k + 2 * hi + 1][colTile * 16 + m];
        acc = __builtin_amdgcn_wmma_f32_16x16x4_f32(
            /*neg_a=*/false, a, /*neg_b=*/false, b,
            /*c_mod=*/(short)0, acc, /*reuse_a=*/false, /*reuse_b=*/false);
    }

    const int rowBase = rowBlock + rowTile * 16;
    const int col = colTile * 16 + m;
    float* hp = H + (size_t)(rowBase + 8 * hi) * 64 + col;
    if (rowBase + 16 <= N) {        // uniform fast path (always taken: N%32==0)
#pragma unroll
        for (int r = 0; r < 8; ++r) hp[(size_t)r * 64] = acc[r];
    } else {
#pragma unroll
        for (int r = 0; r < 8; ++r)
            if (rowBase + 8 * hi + r < N) hp[(size_t)r * 64] = acc[r];
    }
}

// ---------------------------------------------------------------------------
// Init aggregation buffer with self-loop + bias:  agg = h*dinv^2 + b.
// (Replaces zero-fill; edge scatter then atomically accumulates on top.)
// ---------------------------------------------------------------------------
__global__ void init_agg_kernel(float* __restrict__ agg, const float* __restrict__ h,
                                const float* __restrict__ dinv,
                                const float* __restrict__ bias, int N) {
    long long gid = (long long)blockIdx.x * blockDim.x + threadIdx.x;
    if (gid >= (long long)N * 64) return;
    int i = (int)(gid >> 6), c = (int)(gid & 63);
    float di = dinv[i];
    agg[gid] = h[gid] * di * di + bias[c];
}

// ---------------------------------------------------------------------------
// Edge scatter: agg[dst] += h[src] * dinv[src]*dinv[dst].  16 lanes per edge,
// each lane owns 4 contiguous columns (float4 gather + 4 f32 atomics).
// Prefetch the gathered row a few thousand edges ahead (index stream is
// sequential, so the future src index is already in cache).
// ---------------------------------------------------------------------------
__global__ void scatter_kernel(const float* __restrict__ h,
                               const int* __restrict__ src,
                               const int* __restrict__ dst,
                               const float* __restrict__ dinv,
                               float* __restrict__ agg, int E) {
    long long gid = (long long)blockIdx.x * blockDim.x + threadIdx.x;
    int e = (int)(gid >> 4);
    if (e >= E) return;
    int c = ((int)gid & 15) << 2;
    int e2 = e + 4096;
    if (e2 < E) __builtin_prefetch(h + (size_t)src[e2] * 64 + c, 0, 0);
    int s = src[e], d = dst[e];
    float nrm = dinv[s] * dinv[d];
    const float4 hv = *(const float4*)(h + (size_t)s * 64 + c);
    float* o = agg + (size_t)d * 64 + c;
    atomicAdd(o + 0, hv.x * nrm);
    atomicAdd(o + 1, hv.y * nrm);
    atomicAdd(o + 2, hv.z * nrm);
    atomicAdd(o + 3, hv.w * nrm);
}

// ---------------------------------------------------------------------------
// Global mean pool (accumulate), with ReLU fused on read:
//   psum[batch[i]] += relu(x[i]), cnt[batch[i]] += 1.
// ---------------------------------------------------------------------------
__global__ void pool_kernel(const float* __restrict__ x, const int* __restrict__ batch,
                            float* __restrict__ psum, float* __restrict__ cnt, int N) {
    long long gid = (long long)blockIdx.x * blockDim.x + threadIdx.x;
    if (gid >= (long long)N * 64) return;
    int i = (int)(gid >> 6), c = (int)(gid & 63);
    int g = batch[i];
    atomicAdd(&psum[g * 64 + c], fmaxf(x[gid], 0.0f));
    if (c == 0) atomicAdd(&cnt[g], 1.0f);
}

// ---------------------------------------------------------------------------
// Graph branch: g = relu(BN(gf @ Wg + bg)).  64 graphs x 64 channels (tiny).
// ---------------------------------------------------------------------------
__global__ void graph_kernel(const float* __restrict__ gf, const float* __restrict__ Wg,
                             const float* __restrict__ bg, const float* __restrict__ gamma,
                             const float* __restrict__ beta, const float* __restrict__ mean,
                             const float* __restrict__ var, float* __restrict__ gbuf) {
    int b = blockIdx.x, c = threadIdx.x;  // 64 blocks x 64 threads
    float acc = bg[c];
    for (int k = 0; k < 128; ++k) acc += gf[b * 128 + k] * Wg[k * 64 + c];
    acc = (acc - mean[c]) * rsqrtf(var[c] + 1e-5f) * gamma[c] + beta[c];
    gbuf[b * 64 + c] = acc > 0.0f ? acc : 0.0f;
}

// ---------------------------------------------------------------------------
// Classifier head: out = relu([pooled, g] @ Wc1 + bc1) @ Wc2 + bc2.
// One block per graph, 32 threads (= 32 hidden units), LDS reduce.
// Pool finalize (divide by count) folded in here.
// ---------------------------------------------------------------------------
__global__ void head_kernel(const float* __restrict__ psum, const float* __restrict__ cnt,
                            const float* __restrict__ gbuf, const float* __restrict__ Wc1,
                            const float* __restrict__ bc1, const float* __restrict__ Wc2,
                            const float* __restrict__ bc2, float* __restrict__ out) {
    int b = blockIdx.x, j = threadIdx.x;  // 64 blocks x 32 threads
    float invc = 1.0f / fmaxf(cnt[b], 1.0f);
    float acc = bc1[j];
    for (int k = 0; k < 64; ++k) acc += psum[b * 64 + k] * invc * Wc1[k * 32 + j];
    for (int k = 0; k < 64; ++k) acc += gbuf[b * 64 + k] * Wc1[(64 + k) * 32 + j];
    acc = acc > 0.0f ? acc : 0.0f;
    __shared__ float sh[32];
    sh[j] = acc * Wc2[j];
    __syncthreads();
    if (j == 0) {
        float t = 0.0f;
        for (int k = 0; k < 32; ++k) t += sh[k];
        out[b] = t + bc2[0];
    }
}

// ---------------------------------------------------------------------------
// Launch
// ---------------------------------------------------------------------------
extern "C" void kernel_launch(void* const* d_in, const int* in_sizes, int n_in,
                              void* d_out, int out_size, void* d_ws, size_t ws_size,
                              hipStream_t stream) {
    const float* x     = (const float*)d_in[0];
    const int*   ei    = (const int*)d_in[1];   // [2, E]
    const int*   batch = (const int*)d_in[2];   // [N]
    const float* gf    = (const float*)d_in[3];
    const float* Wg1   = (const float*)d_in[4];
    const float* bg1   = (const float*)d_in[5];
    const float* Wg2   = (const float*)d_in[6];
    const float* bg2   = (const float*)d_in[7];
    const float* Wg    = (const float*)d_in[8];
    const float* bgv   = (const float*)d_in[9];
    const float* gamma = (const float*)d_in[10];
    const float* beta  = (const float*)d_in[11];
    const float* mean  = (const float*)d_in[12];
    const float* var   = (const float*)d_in[13];
    const float* Wc1   = (const float*)d_in[14];
    const float* bc1   = (const float*)d_in[15];
    const float* Wc2   = (const float*)d_in[16];
    const float* bc2   = (const float*)d_in[17];
    float* out = (float*)d_out;

    const int N = in_sizes[2];          // 100000
    const int E = in_sizes[1] / 2;      // 3200000
    const int G = 64;
    const int* src = ei;
    const int* dst = ei + E;

    // Workspace carve-up (floats).
    float* dinv = (float*)d_ws;                            // N
    float* bufA = dinv + (((size_t)N + 63) & ~(size_t)63); // N*64 (h)
    float* bufB = bufA + (size_t)N * 64;                   // N*64 (agg / x)
    float* psum = bufB + (size_t)N * 64;                   // G*64
    float* cntb = psum + G * 64;                           // G
    float* gbuf = cntb + 64;                               // G*64

    const long long NH = (long long)N * 64;
    const int bs = 256;
    dim3 blk(bs);
    auto blocks = [](long long n, int b) { return dim3((unsigned)((n + b - 1) / b)); };

    // --- degrees (with self-loop) ---
    fill_kernel<<<blocks(N, bs), blk, 0, stream>>>(dinv, 1.0f, N);
    deg_kernel<<<blocks(E, bs), blk, 0, stream>>>(dst, dinv, E);
    rsqrt_kernel<<<blocks(N, bs), blk, 0, stream>>>(dinv, N);

    // --- layer 1: h1 = X @ W1 ; agg = selfloop+bias ; scatter edges ---
    gemm_wmma_kernel<128, false><<<dim3((N + 31) / 32), blk, 0, stream>>>(x, Wg1, bufA, N);
    init_agg_kernel<<<blocks(NH, bs), blk, 0, stream>>>(bufB, bufA, dinv, bg1, N);
    scatter_kernel<<<blocks((long long)E * 16, bs), blk, 0, stream>>>(bufA, src, dst, dinv, bufB, E);

    // --- layer 2: h2 = relu(pre1) @ W2 (ReLU fused into GEMM staging) ---
    gemm_wmma_kernel<64, true><<<dim3((N + 31) / 32), blk, 0, stream>>>(bufB, Wg2, bufA, N);
    init_agg_kernel<<<blocks(NH, bs), blk, 0, stream>>>(bufB, bufA, dinv, bg2, N);
    scatter_kernel<<<blocks((long long)E * 16, bs), blk, 0, stream>>>(bufA, src, dst, dinv, bufB, E);

    // --- mean pool (ReLU fused on read) ---
    fill_kernel<<<blocks(G * 64 + G, bs), blk, 0, stream>>>(psum, 0.0f, G * 64 + G);
    pool_kernel<<<blocks(NH, bs), blk, 0, stream>>>(bufB, batch, psum, cntb, N);

    // --- graph branch + classifier head ---
    graph_kernel<<<dim3(G), dim3(64), 0, stream>>>(gf, Wg, bgv, gamma, beta, mean, var, gbuf);
    head_kernel<<<dim3(G), dim3(32), 0, stream>>>(psum, cntb, gbuf, Wc1, bc1, Wc2, bc2, out);
}